// GraphConvolutionBlock_45380624449640
// MI455X (gfx1250) — compile-verified
//
#include <hip/hip_runtime.h>

typedef __attribute__((ext_vector_type(2))) float v2f;
typedef __attribute__((ext_vector_type(8))) float v8f;

#define NN   262144   // total nodes
#define EE   524288   // total edges
#define NPER 16384    // nodes per graph
#define KK   8192     // kept per graph
#define BB   16       // graphs
#define DD   128      // feature dim
#define LDSW 136      // padded LDS row stride (2*136 mod 64 == 16 -> conflict-free half-waves)

// ---------------- degree / normalization ----------------
__global__ __launch_bounds__(256) void deg_init_k(float* __restrict__ deg) {
  int i = blockIdx.x * 256 + threadIdx.x;
  if (i < NN) deg[i] = 1.0f;                       // self-loop weight
}

__global__ __launch_bounds__(256) void deg_edge_k(const int* __restrict__ col,
                                                  const float* __restrict__ w,
                                                  float* __restrict__ deg) {
  int e = blockIdx.x * 256 + threadIdx.x;
  if (e < EE) atomicAdd(&deg[col[e]], w[e]);
}

__global__ __launch_bounds__(256) void deg_rsqrt_k(float* __restrict__ deg) {
  int i = blockIdx.x * 256 + threadIdx.x;
  if (i < NN) deg[i] = rsqrtf(deg[i]);             // deg >= 1 always
}

// ---------------- xw = X @ W via V_WMMA_F32_16X16X4_F32 ----------------
// One wave computes a 16-row x 128-col output panel. W staged in LDS (padded).
__global__ __launch_bounds__(256) void gemm_xw_k(const float* __restrict__ X,
                                                 const float* __restrict__ W,
                                                 float* __restrict__ XW) {
  extern __shared__ float lw[];                    // 128 x LDSW floats
  for (int i = threadIdx.x; i < DD * (DD / 4); i += 256) {
    int rr = i >> 5;
    int c4 = (i & 31) << 2;
    const float4 v = *(const float4*)(W + rr * DD + c4);
    float* dst = lw + rr * LDSW + c4;
    dst[0] = v.x; dst[1] = v.y; dst[2] = v.z; dst[3] = v.w;
  }
  __syncthreads();

  const int wave = blockIdx.x * 8 + (threadIdx.x >> 5);
  const int lane = threadIdx.x & 31;
  const int half = lane >> 4;                      // 0: lanes 0-15, 1: lanes 16-31
  const int r    = lane & 15;
  const int m0   = wave << 4;
  if (m0 >= NN) return;

  const float* xrow  = X  + (size_t)(m0 + r) * DD; // A: row m0+r, K along VGPRs
  float*       hbase = XW + (size_t)m0 * DD;

  for (int nt = 0; nt < 8; ++nt) {                 // 8 N-tiles of 16 cols
    v8f acc = {};
    const float* bp = lw + nt * 16 + r;            // B: col 16*nt+r per lane
#pragma unroll
    for (int kb = 0; kb < 32; ++kb) {              // K = 128 as 32 steps of 4
      const int k0 = kb * 4 + half * 2;            // A/B vgpr0 holds K=k0, vgpr1 K=k0+1
      v2f a, b;
      a[0] = xrow[k0];
      a[1] = xrow[k0 + 1];
      b[0] = bp[k0 * LDSW];
      b[1] = bp[(k0 + 1) * LDSW];
      acc = __builtin_amdgcn_wmma_f32_16x16x4_f32(false, a, false, b,
                                                  (short)0, acc, false, false);
    }
    // D layout: VGPR i -> row m0 + i + 8*half, col 16*nt + r
    float* hp = hbase + nt * 16 + r;
#pragma unroll
    for (int i = 0; i < 8; ++i)
      hp[(size_t)(i + 8 * half) * DD] = acc[i];
  }
}

// ---------------- h init: self-loop contribution ----------------
__global__ __launch_bounds__(256) void h_init_k(const float* __restrict__ xw,
                                                const float* __restrict__ dinv,
                                                float* __restrict__ h) {
  int t = blockIdx.x * 256 + threadIdx.x;
  int node = t >> 5, lane = t & 31;
  if (node >= NN) return;
  float d = dinv[node];
  float c = d * d;
  const float4 v = *(const float4*)(xw + (size_t)node * DD + lane * 4);
  float4 o; o.x = c * v.x; o.y = c * v.y; o.z = c * v.z; o.w = c * v.w;
  *(float4*)(h + (size_t)node * DD + lane * 4) = o;
}

// ---------------- edge scatter: h[col] += norm * xw[row] ----------------
__global__ __launch_bounds__(256) void edge_scatter_k(const int* __restrict__ row,
                                                      const int* __restrict__ col,
                                                      const float* __restrict__ w,
                                                      const float* __restrict__ dinv,
                                                      const float* __restrict__ xw,
                                                      float* __restrict__ h) {
  int t = blockIdx.x * 256 + threadIdx.x;
  int e = t >> 5, lane = t & 31;                   // one wave per edge
  if (e >= EE) return;
  int rs = row[e], cs = col[e];
  float coef = dinv[rs] * w[e] * dinv[cs];
  const float4 v = *(const float4*)(xw + (size_t)rs * DD + lane * 4);
  float* hp = h + (size_t)cs * DD + lane * 4;
  atomicAdd(hp + 0, coef * v.x);
  atomicAdd(hp + 1, coef * v.y);
  atomicAdd(hp + 2, coef * v.z);
  atomicAdd(hp + 3, coef * v.w);
}

// ---------------- bias + leaky_relu + pooling score ----------------
__global__ __launch_bounds__(256) void act_score_k(float* __restrict__ h,
                                                   const float* __restrict__ bias,
                                                   const float* __restrict__ pw,
                                                   float* __restrict__ score) {
  int t = blockIdx.x * 256 + threadIdx.x;
  int node = t >> 5, lane = t & 31;
  if (node >= NN) return;
  const float4 b4 = *(const float4*)(bias + lane * 4);
  const float4 p4 = *(const float4*)(pw + lane * 4);
  float* hp = h + (size_t)node * DD + lane * 4;
  float4 v = *(float4*)hp;
  v.x += b4.x; v.y += b4.y; v.z += b4.z; v.w += b4.w;
  v.x = v.x > 0.0f ? v.x : 0.01f * v.x;
  v.y = v.y > 0.0f ? v.y : 0.01f * v.y;
  v.z = v.z > 0.0f ? v.z : 0.01f * v.z;
  v.w = v.w > 0.0f ? v.w : 0.01f * v.w;
  *(float4*)hp = v;
  float dot = v.x * p4.x + v.y * p4.y + v.z * p4.z + v.w * p4.w;
  float psq = p4.x * p4.x + p4.y * p4.y + p4.z * p4.z + p4.w * p4.w;
#pragma unroll
  for (int m = 16; m > 0; m >>= 1) {
    dot += __shfl_xor(dot, m, 32);
    psq += __shfl_xor(psq, m, 32);
  }
  if (lane == 0) score[node] = tanhf(dot * rsqrtf(psq));
}

// ---------------- per-graph top-K by full bitonic sort in LDS ----------------
__global__ __launch_bounds__(1024) void topk_k(const float* __restrict__ score,
                                               int* __restrict__ mapping,
                                               int* __restrict__ perm,
                                               float* __restrict__ gate,
                                               float* __restrict__ out_batch) {
  extern __shared__ char smem[];
  float* key = (float*)smem;                       // NPER floats
  int*   idx = (int*)(smem + NPER * sizeof(float));// NPER ints
  const int b = blockIdx.x;
  const float* s = score + (size_t)b * NPER;
  for (int i = threadIdx.x; i < NPER; i += blockDim.x) { key[i] = s[i]; idx[i] = i; }
  __syncthreads();
  for (int k = 2; k <= NPER; k <<= 1) {
    for (int j = k >> 1; j > 0; j >>= 1) {
      for (int i = threadIdx.x; i < NPER; i += blockDim.x) {
        int l = i ^ j;
        if (l > i) {
          bool desc = ((i & k) == 0);              // final pass -> globally descending
          float ki = key[i], kl = key[l];
          int   ii = idx[i], il = idx[l];
          bool sw = desc ? (ki < kl || (ki == kl && ii > il))
                         : (ki > kl || (ki == kl && ii < il));
          if (sw) { key[i] = kl; key[l] = ki; idx[i] = il; idx[l] = ii; }
        }
      }
      __syncthreads();
    }
  }
  for (int r = threadIdx.x; r < NPER; r += blockDim.x) {
    int g_old = b * NPER + idx[r];
    if (r < KK) {
      int nid = b * KK + r;
      mapping[g_old] = nid;
      perm[nid] = g_old;
      gate[nid] = key[r];
      out_batch[nid] = (float)b;
    } else {
      mapping[g_old] = -1;
    }
  }
}

// ---------------- gather kept nodes, gate by score ----------------
__global__ __launch_bounds__(256) void gather_k(const float* __restrict__ h,
                                                const int* __restrict__ perm,
                                                const float* __restrict__ gate,
                                                float* __restrict__ xout) {
  int t = blockIdx.x * 256 + threadIdx.x;
  int nid = t >> 5, lane = t & 31;
  if (nid >= BB * KK) return;
  int src = perm[nid];
  float g = gate[nid];
  float4 v = *(const float4*)(h + (size_t)src * DD + lane * 4);
  v.x *= g; v.y *= g; v.z *= g; v.w *= g;
  *(float4*)(xout + (size_t)nid * DD + lane * 4) = v;
}

// ---------------- filter_adj: relabel / mask edges ----------------
__global__ __launch_bounds__(256) void filter_k(const int* __restrict__ row,
                                                const int* __restrict__ col,
                                                const float* __restrict__ w,
                                                const int* __restrict__ mapping,
                                                float* __restrict__ out_ei,
                                                float* __restrict__ out_ew) {
  int e = blockIdx.x * 256 + threadIdx.x;
  if (e >= EE) return;
  int r2 = mapping[row[e]];
  int c2 = mapping[col[e]];
  bool valid = (r2 >= 0) && (c2 >= 0);
  out_ei[e]      = valid ? (float)r2 : 0.0f;
  out_ei[EE + e] = valid ? (float)c2 : 0.0f;
  out_ew[e]      = valid ? w[e] : 0.0f;
}

extern "C" void kernel_launch(void* const* d_in, const int* in_sizes, int n_in,
                              void* d_out, int out_size, void* d_ws, size_t ws_size,
                              hipStream_t stream) {
  const float* x  = (const float*)d_in[0];
  const int*   ei = (const int*)d_in[1];
  const float* ew = (const float*)d_in[2];
  // d_in[3] = batch (recomputed directly)
  const float* cw = (const float*)d_in[4];
  const float* cb = (const float*)d_in[5];
  const float* pw = (const float*)d_in[6];
  const int* row = ei;
  const int* col = ei + EE;

  // workspace carve-out
  char* p = (char*)d_ws;
  float* xw    = (float*)p; p += (size_t)NN * DD * sizeof(float);
  float* h     = (float*)p; p += (size_t)NN * DD * sizeof(float);
  float* dinv  = (float*)p; p += (size_t)NN * sizeof(float);      // deg then rsqrt in place
  float* score = (float*)p; p += (size_t)NN * sizeof(float);
  int*   mapping = (int*)p; p += (size_t)NN * sizeof(int);
  int*   perm    = (int*)p; p += (size_t)BB * KK * sizeof(int);
  float* gate    = (float*)p; p += (size_t)BB * KK * sizeof(float);

  // output layout (all emitted as float, ints exact-valued)
  float* out      = (float*)d_out;
  float* out_x    = out;                                 // [BB*KK, DD]
  float* out_ei   = out_x + (size_t)BB * KK * DD;        // [2, EE]
  float* out_ew   = out_ei + (size_t)2 * EE;             // [EE]
  float* out_bat  = out_ew + (size_t)EE;                 // [BB*KK]

  deg_init_k <<<NN / 256, 256, 0, stream>>>(dinv);
  deg_edge_k <<<EE / 256, 256, 0, stream>>>(col, ew, dinv);
  deg_rsqrt_k<<<NN / 256, 256, 0, stream>>>(dinv);

  gemm_xw_k<<<NN / 16 / 8, 256, DD * LDSW * sizeof(float), stream>>>(x, cw, xw);

  h_init_k      <<<NN * 32 / 256, 256, 0, stream>>>(xw, dinv, h);
  edge_scatter_k<<<EE * 32 / 256, 256, 0, stream>>>(row, col, ew, dinv, xw, h);
  act_score_k   <<<NN * 32 / 256, 256, 0, stream>>>(h, cb, pw, score);

  topk_k<<<BB, 1024, NPER * (sizeof(float) + sizeof(int)), stream>>>(
      score, mapping, perm, gate, out_bat);

  gather_k<<<BB * KK * 32 / 256, 256, 0, stream>>>(h, perm, gate, out_x);
  filter_k<<<EE / 256, 256, 0, stream>>>(row, col, ew, mapping, out_ei, out_ew);
}